// RoIPooler_25701084299944
// MI455X (gfx1250) — compile-verified
//
#include <hip/hip_runtime.h>
#include <cstddef>

#define OUTDIM 7
#define NCH 256
#define BINS 49                        // 7*7
#define CGRPS 64                       // channel groups; each thread does 4 channels (stride 64)
#define PER_BOX_THREADS (CGRPS * BINS) // 3136
#define NBOXES 1024
#define BLOCK 224                      // 7 wave32 waves; 3136 / 224 = 14 exact
#define BLOCKS_PER_BOX (PER_BOX_THREADS / BLOCK) // 14

__global__ __launch_bounds__(BLOCK)
void roi_align_fpn_kernel(const float* __restrict__ fm2,
                          const float* __restrict__ fm3,
                          const float* __restrict__ fm4,
                          const float* __restrict__ fm5,
                          const float* __restrict__ boxes1,
                          const float* __restrict__ boxes2,
                          float* __restrict__ out)
{
    // Block-uniform box id -> box math scalarizes (s_load_b128 + SALU select).
    const int blk  = blockIdx.x;
    const int m    = blk / BLOCKS_PER_BOX;                     // box id, SGPR
    const int r    = (blk - m * BLOCKS_PER_BOX) * BLOCK + threadIdx.x; // 0..3135
    const int cgrp = r / BINS;                                 // channel group 0..63
    const int bin  = r - cgrp * BINS;                          // 0..48 (fastest -> coalesced stores)
    const int oy   = bin / OUTDIM;
    const int ox   = bin - oy * OUTDIM;

    // Box fetch: boxes1 -> batch 0, boxes2 -> batch 1 (concatenated order).
    // Address is wave-uniform -> scalar load path.
    const float* bx;
    int batch;
    if (m < 512) { bx = boxes1 + 4 * m;         batch = 0; }
    else         { bx = boxes2 + 4 * (m - 512); batch = 1; }
    const float x1 = bx[0], y1 = bx[1], x2 = bx[2], y2 = bx[3];

    // FPN level: floor(log2(sqrt(area)/224) + 4), clipped to [2,5].
    // 0.5*log2(area) + (4 - log2(224)); log2(224) = 7.807354922057604
    // area >= ~1 (boxes clamped to >=1px), so hardware v_log_f32 is safe.
    const float area = (x2 - x1) * (y2 - y1);
    const float lvf  = floorf(0.5f * __log2f(area) - 3.8073549220576041f);
    int lvl = (int)lvf;
    lvl = lvl < 2 ? 2 : (lvl > 5 ? 5 : lvl);

    // Level -> feature map (all levels are square: H == W == S). Block-uniform branch.
    const float* fm = fm2; int S = 200; float invs = 0.25f;       // stride 4
    if      (lvl == 3) { fm = fm3; S = 100; invs = 0.125f;   }    // stride 8
    else if (lvl == 4) { fm = fm4; S = 50;  invs = 0.0625f;  }    // stride 16
    else if (lvl == 5) { fm = fm5; S = 25;  invs = 0.03125f; }    // stride 32

    // Single-sample bilinear position for this bin (matches reference exactly).
    const float x1s = x1 * invs, y1s = y1 * invs;
    const float bw  = (x2 - x1) * invs * (1.0f / OUTDIM);
    const float bh  = (y2 - y1) * invs * (1.0f / OUTDIM);
    const float lim = (float)(S - 1);

    float gx = x1s + ((float)ox + 0.5f) * bw;
    float gy = y1s + ((float)oy + 0.5f) * bh;
    gx = fminf(fmaxf(gx, 0.0f), lim);
    gy = fminf(fmaxf(gy, 0.0f), lim);

    const float x0f = floorf(gx), y0f = floorf(gy);
    const int ix0 = (int)x0f, iy0 = (int)y0f;
    const int ix1 = min(ix0 + 1, S - 1);
    const int iy1 = min(iy0 + 1, S - 1);
    const float wx = gx - x0f, wy = gy - y0f;
    const float w00 = (1.0f - wy) * (1.0f - wx);
    const float w01 = (1.0f - wy) * wx;
    const float w10 = wy * (1.0f - wx);
    const float w11 = wy * wx;

    const size_t HW  = (size_t)S * (size_t)S;
    const int o00 = iy0 * S + ix0;
    const int o01 = iy0 * S + ix1;
    const int o10 = iy1 * S + ix0;
    const int o11 = iy1 * S + ix1;

    // 4 channels per thread: c = cgrp + 64*k. Weights computed once, reused.
    const float* plane     = fm + ((size_t)batch * NCH + (size_t)cgrp) * HW;
    const size_t outbase   = (size_t)m * (NCH * BINS) + (size_t)cgrp * BINS + (size_t)bin;
    const size_t planeStep = (size_t)CGRPS * HW;    // next channel group's plane
    const size_t outStep   = (size_t)CGRPS * BINS;

    #pragma unroll
    for (int k = 0; k < 4; ++k) {
        const float* p = plane + (size_t)k * planeStep;
        if (k < 3) {
            // CDNA5 global_prefetch_b8: pre-touch next plane's corner cachelines.
            const float* pn = p + planeStep;
            __builtin_prefetch(pn + o00, 0, 1);
            __builtin_prefetch(pn + o10, 0, 1);
        }
        const float v00 = p[o00];
        const float v01 = p[o01];
        const float v10 = p[o10];
        const float v11 = p[o11];
        out[outbase + (size_t)k * outStep] =
            v00 * w00 + v01 * w01 + v10 * w10 + v11 * w11;
    }
}

extern "C" void kernel_launch(void* const* d_in, const int* in_sizes, int n_in,
                              void* d_out, int out_size, void* d_ws, size_t ws_size,
                              hipStream_t stream) {
    (void)in_sizes; (void)n_in; (void)out_size; (void)d_ws; (void)ws_size;
    const float* fm2    = (const float*)d_in[0];
    const float* fm3    = (const float*)d_in[1];
    const float* fm4    = (const float*)d_in[2];
    const float* fm5    = (const float*)d_in[3];
    const float* boxes1 = (const float*)d_in[4];
    const float* boxes2 = (const float*)d_in[5];
    float* out = (float*)d_out;

    const int grid = NBOXES * BLOCKS_PER_BOX;   // 14,336 blocks, 224 threads each
    roi_align_fpn_kernel<<<grid, BLOCK, 0, stream>>>(fm2, fm3, fm4, fm5,
                                                     boxes1, boxes2, out);
}